// PosePostModel_14637248545309
// MI455X (gfx1250) — compile-verified
//
#include <hip/hip_runtime.h>
#include <stdint.h>

// Problem constants (from reference)
#define NB 16          // batch
#define NH 320         // height
#define NW 320         // width
#define NP 64          // params per peak
#define TOPK 100
#define THRES 0.5f

// Phase-1 tiling
#define RPW 32             // rows produced per workgroup
#define LROWS (RPW + 2)    // + halo rows

// d_out layout (floats, reference return order)
#define OFF_COORS  0
#define OFF_PARAMS (NB * TOPK * 2)                 // 3200
#define OFF_SCORES (OFF_PARAMS + NB * TOPK * NP)   // 105600
#define OFF_MASK   (OFF_SCORES + NB * TOPK)        // 107200

typedef unsigned long long u64;

__device__ __forceinline__ u64 wave_max32(u64 v) {
    // wave32 butterfly max via lane shuffles (no LDS, no barriers)
    #pragma unroll
    for (int m = 16; m > 0; m >>= 1) {
        u64 o = __shfl_xor(v, m, 32);
        if (o > v) v = o;
    }
    return v;
}

__global__ void zero_counters_kernel(int* __restrict__ cnt) {
    if (threadIdx.x < NB) cnt[threadIdx.x] = 0;
}

// ---------------------------------------------------------------------------
// Phase 1: 3x3 peak-NMS + threshold + candidate compaction.
// Heatmap rows are staged into LDS with CDNA5 async global->LDS DMA
// (GLOBAL_LOAD_ASYNC_TO_LDS_B128, tracked by ASYNCcnt).
// ---------------------------------------------------------------------------
__global__ __launch_bounds__(256) void peaks_kernel(const float* __restrict__ hms,
                                                    int* __restrict__ cnt,
                                                    u64* __restrict__ cand,
                                                    int cap) {
    __shared__ __align__(16) float smem[LROWS * NW];

    const int b   = blockIdx.y;
    const int r0  = blockIdx.x * RPW;
    const int tid = threadIdx.x;

    // --- stage LROWS rows into LDS: 16 bytes per lane per async op ---
    const int nq = LROWS * (NW / 4);   // float4 elements
    for (int i = tid; i < nq; i += 256) {
        const int li = i / (NW / 4);
        const int q  = i % (NW / 4);
        const int g  = r0 - 1 + li;    // global row (halo offset)
        if (g >= 0 && g < NH) {
            // Low 32 bits of a flat pointer to __shared__ are the LDS byte offset.
            unsigned lds_addr = (unsigned)(uintptr_t)(&smem[li * NW + q * 4]);
            u64 gaddr = (u64)(uintptr_t)(hms + ((size_t)(b * NH + g)) * NW + (size_t)q * 4);
            asm volatile("global_load_async_to_lds_b128 %0, %1, off"
                         :: "v"(lds_addr), "v"(gaddr)
                         : "memory");
        } else {
            // SAME padding: out-of-range rows behave as -inf
            float4 m = make_float4(-1e30f, -1e30f, -1e30f, -1e30f);
            *(float4*)(&smem[li * NW + q * 4]) = m;
        }
    }
    asm volatile("s_wait_asynccnt 0" ::: "memory");
    __syncthreads();

    // --- peak test: center >= all in-range 3x3 neighbors, value > 0.5 ---
    for (int i = tid; i < RPW * NW; i += 256) {
        const int dy = i / NW;
        const int x  = i % NW;
        const int li = dy + 1;
        const float v = smem[li * NW + x];
        if (v > THRES) {
            bool peak = true;
            if (x > 0)      peak &= (v >= smem[li * NW + x - 1]);
            if (x < NW - 1) peak &= (v >= smem[li * NW + x + 1]);
            const int xm = (x > 0) ? x - 1 : x;
            const int xp = (x < NW - 1) ? x + 1 : x;
            #pragma unroll
            for (int d = -1; d <= 1; d += 2) {
                const int lr = li + d;   // halo rows hold -inf when OOB
                peak &= (v >= smem[lr * NW + xm]);
                peak &= (v >= smem[lr * NW + x ]);
                peak &= (v >= smem[lr * NW + xp]);
            }
            if (peak) {
                const unsigned idx = (unsigned)((r0 + dy) * NW + x);
                // key: value bits (positive floats order like uints) then ~idx
                // so ties resolve to the smaller flat index, matching lax.top_k.
                const u64 key = ((u64)__float_as_uint(v) << 32) |
                                (u64)(0xFFFFFFFFu - idx);
                const int pos = atomicAdd(&cnt[b], 1);
                if (pos < cap) cand[(size_t)b * cap + pos] = key;
            }
        }
    }
}

// ---------------------------------------------------------------------------
// Phase 2: per-batch top-100 (descending, index tie-break) + output writes.
// One 1024-thread workgroup (32 wave32s) per batch. Per round: shuffle-based
// intra-wave max, then a single 32-entry LDS stage reduced by wave 0.
// ---------------------------------------------------------------------------
__global__ __launch_bounds__(1024) void topk_kernel(const float* __restrict__ pms,
                                                    const int* __restrict__ cnt,
                                                    u64* __restrict__ cand,
                                                    int cap,
                                                    float* __restrict__ out) {
    __shared__ u64 red[32];      // one slot per wave
    __shared__ u64 sel[TOPK];

    const int b    = blockIdx.x;
    const int tid  = threadIdx.x;
    const int lane = tid & 31;
    const int wid  = tid >> 5;   // 0..31

    int n = cnt[b];
    if (n > cap) n = cap;
    u64* __restrict__ cb = cand + (size_t)b * cap;

    if (tid < TOPK) sel[tid] = 0ull;
    __syncthreads();

    u64 prev = 0ull;
    for (int r = 0; r < TOPK; ++r) {
        u64 best = 0ull;
        for (int j = tid; j < n; j += 1024) {
            u64 k = cb[j];
            if (prev != 0ull && k == prev) { cb[j] = 0ull; k = 0ull; }  // retire winner
            if (k > best) best = k;
        }
        best = wave_max32(best);
        if (lane == 0) red[wid] = best;
        __syncthreads();
        if (wid == 0) {
            u64 w = red[lane];
            w = wave_max32(w);
            if (lane == 0) red[0] = w;
        }
        __syncthreads();
        const u64 win = red[0];
        __syncthreads();                    // protect red[] before next round
        if (win == 0ull) break;             // uniform: no candidates left
        if (tid == 0) sel[r] = win;
        prev = win;
    }
    __syncthreads();

    // --- coors / scores / mask ---
    if (tid < TOPK) {
        const u64 k = sel[tid];
        const bool valid = (k != 0ull);
        const unsigned idx = valid ? (0xFFFFFFFFu - (unsigned)(k & 0xFFFFFFFFull)) : 0u;
        const float sc = valid ? __uint_as_float((unsigned)(k >> 32)) : 0.0f;
        const int y = (int)(idx / NW);
        const int x = (int)(idx % NW);
        const size_t o = (size_t)(b * TOPK + tid);
        out[OFF_COORS + o * 2 + 0] = valid ? (float)y : 0.0f;
        out[OFF_COORS + o * 2 + 1] = valid ? (float)x : 0.0f;
        out[OFF_SCORES + o] = sc;
        out[OFF_MASK + o]   = valid ? 1.0f : 0.0f;
    }
    __syncthreads();

    // --- params gather: 100 peaks x 64 floats, float4-vectorized ---
    for (int t = tid; t < TOPK * (NP / 4); t += 1024) {
        const int p = t >> 4;       // peak slot (NP/4 == 16)
        const int q = t & 15;       // float4 index within the 64 params
        const u64 k = sel[p];
        float4 val = make_float4(0.0f, 0.0f, 0.0f, 0.0f);
        if (k != 0ull) {
            const unsigned idx = 0xFFFFFFFFu - (unsigned)(k & 0xFFFFFFFFull);
            const float4* row =
                (const float4*)(pms + ((size_t)b * NH * NW + idx) * NP);
            val = row[q];
        }
        ((float4*)(out + OFF_PARAMS))[(size_t)(b * TOPK + p) * (NP / 4) + q] = val;
    }
}

extern "C" void kernel_launch(void* const* d_in, const int* in_sizes, int n_in,
                              void* d_out, int out_size, void* d_ws, size_t ws_size,
                              hipStream_t stream) {
    const float* hms = (const float*)d_in[0];   // [16,320,320,1] f32
    const float* pms = (const float*)d_in[1];   // [16,320,320,64] f32
    float* out = (float*)d_out;                 // 108800 floats

    // workspace: [0,64) = 16 candidate counters; rest = per-batch u64 key arrays
    int* cnt  = (int*)d_ws;
    u64* cand = (u64*)((char*)d_ws + 64);
    size_t avail = (ws_size > 64) ? (ws_size - 64) : 0;
    int cap = (int)(avail / (sizeof(u64) * NB));
    if (cap > 16384) cap = 16384;   // ~11.4K expected peaks/batch
    if (cap < 1)     cap = 1;

    zero_counters_kernel<<<1, 64, 0, stream>>>(cnt);

    dim3 g1(NH / RPW, NB);          // 10 x 16 tiles
    peaks_kernel<<<g1, 256, 0, stream>>>(hms, cnt, cand, cap);

    topk_kernel<<<NB, 1024, 0, stream>>>(pms, cnt, cand, cap, out);
}